// DistributedSigLoss_58935541236016
// MI455X (gfx1250) — compile-verified
//
#include <hip/hip_runtime.h>
#include <math.h>

typedef __attribute__((ext_vector_type(2))) float v2f;
typedef __attribute__((ext_vector_type(8))) float v8f;

#define DIM 256      // feature dim D
#define BM  64       // block tile rows
#define BN  256      // block tile cols
// 8 waves per block: wave grid 2 (M) x 4 (N); each wave -> 32x64 = 2x4 tiles of 16x16

__global__ void zero_acc_kernel(double* acc) {
    if (threadIdx.x == 0) *acc = 0.0;
}

__global__ __launch_bounds__(256) void normalize_rows_kernel(
    const float* __restrict__ img, const float* __restrict__ txt,
    float* __restrict__ outA, float* __restrict__ outB, int N)
{
    const int lane = threadIdx.x & 31;
    const int wid  = threadIdx.x >> 5;
    const int row  = blockIdx.x * 8 + wid;
    if (row >= 2 * N) return;

    const float* src;
    float* dst;
    if (row < N) { src = img + (size_t)row * DIM;       dst = outA + (size_t)row * DIM; }
    else         { src = txt + (size_t)(row - N) * DIM; dst = outB + (size_t)(row - N) * DIM; }

    // each lane handles 8 consecutive floats (32 lanes * 8 = 256 = DIM)
    float x[8];
    #pragma unroll
    for (int i = 0; i < 8; ++i) x[i] = src[lane * 8 + i];

    float s = 0.0f;
    #pragma unroll
    for (int i = 0; i < 8; ++i) s = fmaf(x[i], x[i], s);

    #pragma unroll
    for (int off = 16; off > 0; off >>= 1) s += __shfl_xor(s, off, 32);

    const float inv = 1.0f / fmaxf(sqrtf(s), 1e-12f);

    #pragma unroll
    for (int i = 0; i < 8; ++i) dst[lane * 8 + i] = x[i] * inv;
}

struct Frags { v2f a0, a1, b0, b1, b2, b3; };

// Tile strides are compile-time constants -> fold into the 24-bit instruction
// offset of global_load_b64 (16*DIM*4 = 16KB), keeping ONE VGPR address per base.
__device__ __forceinline__ Frags load_frags(const float* __restrict__ ap,
                                            const float* __restrict__ bp,
                                            int kk)
{
    Frags f;
    f.a0 = *(const v2f*)(ap + kk);
    f.a1 = *(const v2f*)(ap + kk + 16 * DIM);
    f.b0 = *(const v2f*)(bp + kk);
    f.b1 = *(const v2f*)(bp + kk + 16 * DIM);
    f.b2 = *(const v2f*)(bp + kk + 32 * DIM);
    f.b3 = *(const v2f*)(bp + kk + 48 * DIM);
    return f;
}

__global__ __launch_bounds__(256) void siglip_gemm_loss_kernel(
    const float* __restrict__ An, const float* __restrict__ Bn,
    const float* __restrict__ tprime, const float* __restrict__ biasp,
    double* __restrict__ acc, int N)
{
    __shared__ float red[8];

    const int tid    = threadIdx.x;
    const int lane   = tid & 31;
    const int wid    = tid >> 5;
    const int wave_m = wid >> 2;   // 0..1
    const int wave_n = wid & 3;    // 0..3
    const int half   = lane >> 4;  // 0 or 1 (upper half-wave)
    const int l16    = lane & 15;

    const int blockM = blockIdx.y * BM;
    const int blockN = blockIdx.x * BN;

    // f32 WMMA 16x16x4 fragment layout:
    //  A (16x4): lane L<16 holds row M=L, v0=K0,v1=K1; lane L>=16 holds row M=L-16, v0=K2,v1=K3
    //  B (4x16): symmetric, lane indexes column N, half selects K pair
    //  -> per-lane fragment is a contiguous float2 at [row_or_col * D + kk + 2*half]
    const int rowA0 = blockM + wave_m * 32 + l16;
    const int colB0 = blockN + wave_n * 64 + l16;

    const float* ap = An + (size_t)rowA0 * DIM + 2 * half;
    const float* bp = Bn + (size_t)colB0 * DIM + 2 * half;

    v8f acc_t[2][4] = {};

    // Software pipeline: prefetch next k-step's 6 fragments while 8 WMMAs
    // consume the current set; (kk+4)&255 wraps so the loop is branch-free
    // (last prefetch harmlessly reloads kk=0).
    Frags cur = load_frags(ap, bp, 0);

    #pragma unroll 2
    for (int kk = 0; kk < DIM; kk += 4) {
        Frags nxt = load_frags(ap, bp, (kk + 4) & (DIM - 1));

        acc_t[0][0] = __builtin_amdgcn_wmma_f32_16x16x4_f32(false, cur.a0, false, cur.b0, (short)0, acc_t[0][0], false, false);
        acc_t[0][1] = __builtin_amdgcn_wmma_f32_16x16x4_f32(false, cur.a0, false, cur.b1, (short)0, acc_t[0][1], false, false);
        acc_t[0][2] = __builtin_amdgcn_wmma_f32_16x16x4_f32(false, cur.a0, false, cur.b2, (short)0, acc_t[0][2], false, false);
        acc_t[0][3] = __builtin_amdgcn_wmma_f32_16x16x4_f32(false, cur.a0, false, cur.b3, (short)0, acc_t[0][3], false, false);
        acc_t[1][0] = __builtin_amdgcn_wmma_f32_16x16x4_f32(false, cur.a1, false, cur.b0, (short)0, acc_t[1][0], false, false);
        acc_t[1][1] = __builtin_amdgcn_wmma_f32_16x16x4_f32(false, cur.a1, false, cur.b1, (short)0, acc_t[1][1], false, false);
        acc_t[1][2] = __builtin_amdgcn_wmma_f32_16x16x4_f32(false, cur.a1, false, cur.b2, (short)0, acc_t[1][2], false, false);
        acc_t[1][3] = __builtin_amdgcn_wmma_f32_16x16x4_f32(false, cur.a1, false, cur.b3, (short)0, acc_t[1][3], false, false);

        cur = nxt;
    }

    // Fused SigLIP epilogue: logit = exp(t')*c + b ; z = (M==N)? logit : -logit
    // log_sigmoid(z) = min(z,0) - log1p(exp(-|z|))  (numerically stable)
    const float sc = expf(tprime[0]);
    const float bi = biasp[0];

    float lsum = 0.0f;
    #pragma unroll
    for (int i = 0; i < 2; ++i) {
        #pragma unroll
        for (int j = 0; j < 4; ++j) {
            // C/D layout: VGPR r: lanes 0-15 -> M=r, lanes 16-31 -> M=r+8; N = lane&15
            const int Mb = blockM + wave_m * 32 + i * 16 + half * 8;
            const int Nc = blockN + wave_n * 64 + j * 16 + l16;
            #pragma unroll
            for (int r = 0; r < 8; ++r) {
                const float c = acc_t[i][j][r];
                const float logit = fmaf(sc, c, bi);
                const float z = ((Mb + r) == Nc) ? logit : -logit;
                const float ls = fminf(z, 0.0f) - log1pf(expf(-fabsf(z)));
                lsum += ls;
            }
        }
    }

    // wave32 reduction
    #pragma unroll
    for (int off = 16; off > 0; off >>= 1) lsum += __shfl_xor(lsum, off, 32);
    if (lane == 0) red[wid] = lsum;
    __syncthreads();

    if (wid == 0) {
        float v = (lane < 8) ? red[lane] : 0.0f;
        #pragma unroll
        for (int off = 4; off > 0; off >>= 1) v += __shfl_xor(v, off, 32);
        if (lane == 0) atomicAdd(acc, (double)v);  // global_atomic_add_f64
    }
}

__global__ void finalize_kernel(const double* acc, float* out, int N) {
    if (threadIdx.x == 0) out[0] = (float)(-(*acc) / (double)N);
}

extern "C" void kernel_launch(void* const* d_in, const int* in_sizes, int n_in,
                              void* d_out, int out_size, void* d_ws, size_t ws_size,
                              hipStream_t stream) {
    const float* img = (const float*)d_in[0];
    const float* txt = (const float*)d_in[1];
    const float* tpr = (const float*)d_in[2];
    const float* bia = (const float*)d_in[3];
    const int N = in_sizes[0] / DIM;   // 16384

    float*  An  = (float*)d_ws;
    float*  Bn  = An + (size_t)N * DIM;
    double* acc = (double*)(Bn + (size_t)N * DIM);   // 2*N*DIM*4 bytes is 8B-aligned

    zero_acc_kernel<<<1, 1, 0, stream>>>(acc);

    const int rows2 = 2 * N;
    normalize_rows_kernel<<<(rows2 + 7) / 8, 256, 0, stream>>>(img, txt, An, Bn, N);

    dim3 grid(N / BN, N / BM);   // 64 x 256 blocks for N=16384
    siglip_gemm_loss_kernel<<<grid, 256, 0, stream>>>(An, Bn, tpr, bia, acc, N);

    finalize_kernel<<<1, 1, 0, stream>>>(acc, (float*)d_out, N);
}